// LocallyLowRank_62165356642664
// MI455X (gfx1250) — compile-verified
//
#include <hip/hip_runtime.h>
#include <cmath>

#define N_IMG   4
#define A_CH    16
#define H_DIM   384
#define W_DIM   384
#define HW      (H_DIM * W_DIM)
#define B_SIZE  8
#define STRIDE  4
#define THRESH  2.0f
#define NB      95            // (384-8)/4 + 1
#define NBLK    (NB * NB)     // 9025
#define P_PIX   64

typedef float v2f __attribute__((ext_vector_type(2)));
typedef float v8f __attribute__((ext_vector_type(8)));

__device__ __forceinline__ int pix_index(int bh, int bw, int p) {
    int r = p >> 3, c = p & 7;
    return (bh * STRIDE + r) * W_DIM + (bw * STRIDE + c);
}

__global__ void llr_zero_kernel(float* __restrict__ out, int n) {
    int i = blockIdx.x * blockDim.x + threadIdx.x;
    int stride = gridDim.x * blockDim.x;
    for (; i < n; i += stride) out[i] = 0.0f;
}

// One wave32 per 8x8 block. Gram -> Jacobi eigensolve -> shrink -> reconstruct.
__global__ __launch_bounds__(32)
void llr_block_kernel(const float* __restrict__ x, float* __restrict__ accum) {
    __shared__ float m[P_PIX][A_CH + 1];      // 64 x 17 (pad kills bank conflicts)
    __shared__ float G[A_CH][A_CH + 1];       // Gram; later reused to hold W = V d V^T
    __shared__ float V[A_CH][A_CH + 1];       // eigenvectors
    __shared__ float cs_arr[8], sn_arr[8];
    __shared__ float dscale[A_CH];

    const int lane = threadIdx.x;             // 0..31
    const int lo   = lane & 15;
    const int hi   = lane >> 4;

    const int b  = blockIdx.x;
    const int n  = b / NBLK;
    const int rb = b % NBLK;
    const int bh = rb / NB;
    const int bw = rb % NB;

    // ---- gather m: 64 pixels x 16 channels into LDS -------------------------
    for (int c = 0; c < A_CH; ++c) {
        const float* xc = x + (size_t)(n * A_CH + c) * HW;
        m[lane][c]      = xc[pix_index(bh, bw, lane)];
        m[lane + 32][c] = xc[pix_index(bh, bw, lane + 32)];
    }
    __syncthreads();

    // ---- G = m^T m : 16 chained v_wmma_f32_16x16x4_f32 ----------------------
    // A-frag of m^T equals B-frag of m per lane: {m[k0][lo], m[k0+1][lo]}.
    v8f acc = {};
    for (int kc = 0; kc < 16; ++kc) {
        int k0 = 4 * kc + 2 * hi;
        v2f a;
        a.x = m[k0][lo];
        a.y = m[k0 + 1][lo];
        acc = __builtin_amdgcn_wmma_f32_16x16x4_f32(false, a, false, a,
                                                    (short)0, acc, false, false);
    }
    for (int j = 0; j < 8; ++j) {
        int row = j + 8 * hi;
        G[row][lo] = acc[j];
        V[row][lo] = (row == lo) ? 1.0f : 0.0f;
    }
    __syncthreads();

    // ---- parallel cyclic Jacobi on 16x16 G (tournament: 15 rounds/sweep) ----
    for (int sweep = 0; sweep < 8; ++sweep) {
        for (int r = 0; r < 15; ++r) {
            if (lane < 8) {                    // 8 disjoint pairs -> angles
                int j = lane;
                int p = (j == 0) ? 0 : ((j - 1 + r) % 15) + 1;
                int q = ((14 - j + r) % 15) + 1;
                if (p > q) { int t = p; p = q; q = t; }
                float app = G[p][p], aqq = G[q][q], apq = G[p][q];
                float c = 1.0f, s = 0.0f;
                if (fabsf(apq) > 1e-12f) {
                    float tau = (aqq - app) / (2.0f * apq);
                    float t = ((tau >= 0.0f) ? 1.0f : -1.0f) /
                              (fabsf(tau) + sqrtf(1.0f + tau * tau));
                    c = rsqrtf(1.0f + t * t);
                    s = t * c;
                }
                cs_arr[j] = c; sn_arr[j] = s;
            }
            __syncthreads();
            // column rotations: G <- G J, V <- V J   (128 tasks / 32 lanes)
            for (int t4 = 0; t4 < 4; ++t4) {
                int task = lane + 32 * t4;
                int j = task & 7, i = task >> 3;
                int p = (j == 0) ? 0 : ((j - 1 + r) % 15) + 1;
                int q = ((14 - j + r) % 15) + 1;
                if (p > q) { int t = p; p = q; q = t; }
                float c = cs_arr[j], s = sn_arr[j];
                float gip = G[i][p], giq = G[i][q];
                G[i][p] = c * gip - s * giq;
                G[i][q] = s * gip + c * giq;
                float vip = V[i][p], viq = V[i][q];
                V[i][p] = c * vip - s * viq;
                V[i][q] = s * vip + c * viq;
            }
            __syncthreads();
            // row rotations: G <- J^T G
            for (int t4 = 0; t4 < 4; ++t4) {
                int task = lane + 32 * t4;
                int j = task & 7, i = task >> 3;
                int p = (j == 0) ? 0 : ((j - 1 + r) % 15) + 1;
                int q = ((14 - j + r) % 15) + 1;
                if (p > q) { int t = p; p = q; q = t; }
                float c = cs_arr[j], s = sn_arr[j];
                float gpi = G[p][i], gqi = G[q][i];
                G[p][i] = c * gpi - s * gqi;
                G[q][i] = s * gpi + c * gqi;
            }
            __syncthreads();
        }
    }

    // ---- singular value shrink: d_k = max(sqrt(l)-T,0)/sqrt(l) --------------
    if (lane < 16) {
        float lam = fmaxf(G[lane][lane], 0.0f);
        float sig = sqrtf(lam);
        float sh  = fmaxf(sig - THRESH, 0.0f);
        dscale[lane] = (sig > 1e-20f) ? (sh / sig) : 0.0f;
    }
    __syncthreads();

    // ---- W = (V diag(d)) V^T : 4 WMMAs; result overwrites G -----------------
    v8f accW = {};
    for (int kc = 0; kc < 4; ++kc) {
        int k0 = 4 * kc + 2 * hi;
        v2f a, bf;
        bf.x = V[lo][k0];                      // B = V^T  -> B[k][n] = V[n][k]
        bf.y = V[lo][k0 + 1];
        a.x  = bf.x * dscale[k0];              // A = V diag(d)
        a.y  = bf.y * dscale[k0 + 1];
        accW = __builtin_amdgcn_wmma_f32_16x16x4_f32(false, a, false, bf,
                                                     (short)0, accW, false, false);
    }
    __syncthreads();                           // all reads of G/V done
    for (int j = 0; j < 8; ++j) G[j + 8 * hi][lo] = accW[j];
    __syncthreads();

    // ---- m' = m W : four 16x16 tiles, 4 WMMAs each; scatter-add -------------
    for (int T = 0; T < 4; ++T) {
        v8f accO = {};
        for (int kc = 0; kc < 4; ++kc) {
            int k0 = 4 * kc + 2 * hi;
            v2f a, bf;
            a.x  = m[16 * T + lo][k0];         // A = m tile (pixel x channel)
            a.y  = m[16 * T + lo][k0 + 1];
            bf.x = G[k0][lo];                  // B = W
            bf.y = G[k0 + 1][lo];
            accO = __builtin_amdgcn_wmma_f32_16x16x4_f32(false, a, false, bf,
                                                         (short)0, accO, false, false);
        }
        for (int j = 0; j < 8; ++j) {
            int p = 16 * T + j + 8 * hi;       // pixel index in block
            float* dst = accum + (size_t)(n * A_CH + lo) * HW + pix_index(bh, bw, p);
            atomicAdd(dst, accO[j]);
        }
    }
}

__global__ void llr_normalize_kernel(float* __restrict__ out,
                                     const float* __restrict__ bw, int total) {
    int i = blockIdx.x * blockDim.x + threadIdx.x;
    int stride = gridDim.x * blockDim.x;
    for (; i < total; i += stride) out[i] = out[i] / bw[i % HW];
}

extern "C" void kernel_launch(void* const* d_in, const int* in_sizes, int n_in,
                              void* d_out, int out_size, void* d_ws, size_t ws_size,
                              hipStream_t stream) {
    (void)in_sizes; (void)n_in; (void)d_ws; (void)ws_size; (void)out_size;
    const float* x  = (const float*)d_in[0];
    const float* bw = (const float*)d_in[1];
    float* out = (float*)d_out;
    const int total = N_IMG * A_CH * HW;

    llr_zero_kernel<<<1024, 256, 0, stream>>>(out, total);
    llr_block_kernel<<<N_IMG * NBLK, 32, 0, stream>>>(x, out);
    llr_normalize_kernel<<<1024, 256, 0, stream>>>(out, bw, total);
}